// EarthMoverDistance_26336739459365
// MI455X (gfx1250) — compile-verified
//
#include <hip/hip_runtime.h>
#include <cmath>

// ---------------------------------------------------------------------------
// Entropic-OT (Sinkhorn) approximate EMD for B=8, N=M=2048, dim=3.
// C_ij = |p_i|^2 + |q_j|^2 - 2 p_i.q_j ; the dot-product term is computed with
// V_WMMA_F32_16X16X4_F32 (K=3 padded to 4). Dual updates use a two-pass
// (exact-max, then exp2-sum) logsumexp: pass A is WMMA + v_max only, pass B
// spends exactly one v_exp_f32 per matrix element (exp2 domain, no premul).
// ---------------------------------------------------------------------------

#define BATCH     8
#define NPTS      2048                      // N == M
#define EPS_F     0.02f
#define INV_EPS   (1.0f / EPS_F)            // 50
#define N_ITERS   20

#define LOG2E_F   1.44269504088896340736f
#define LN2_F     0.69314718055994530942f
#define SC2       (2.0f / EPS_F * LOG2E_F)  // dot coefficient in exp2 domain

#define TBLOCK         128                  // 4 waves (wave32)
#define ROWS_PER_WAVE  16
#define ROWS_PER_BLOCK 64                   // 4 waves * 16 rows

typedef float v2f __attribute__((ext_vector_type(2)));
typedef float v8f __attribute__((ext_vector_type(8)));

__device__ __forceinline__ float dev_exp2(float x) {
  return __builtin_amdgcn_exp2f(x);         // v_exp_f32 (hardware exp2)
}

// ---------------------------------------------------------------------------
// Init: squared norms, zero duals, zero output.
// ---------------------------------------------------------------------------
__global__ void emd_init(const float* __restrict__ p1, const float* __restrict__ p2,
                         float* __restrict__ normP, float* __restrict__ normQ,
                         float* __restrict__ f, float* __restrict__ g,
                         float* __restrict__ out) {
  int idx = blockIdx.x * blockDim.x + threadIdx.x;
  if (idx < BATCH * NPTS) {
    float x = p1[idx * 3 + 0], y = p1[idx * 3 + 1], z = p1[idx * 3 + 2];
    normP[idx] = x * x + y * y + z * z;
    x = p2[idx * 3 + 0]; y = p2[idx * 3 + 1]; z = p2[idx * 3 + 2];
    normQ[idx] = x * x + y * y + z * z;
    f[idx] = 0.0f;
    g[idx] = 0.0f;
  }
  if (blockIdx.x == 0 && threadIdx.x < BATCH) out[threadIdx.x] = 0.0f;
}

// ---------------------------------------------------------------------------
// One Sinkhorn half-step:
//   dualOut_i = normP_i - eps*ln2 * ( m2_i + log2( sum_j 2^(x2_ij - m2_i) ) )
//   x2_ij = t2_j + SC2 * dot_ij,  t2_j = ((dualIn_j - normQ_j)/eps + log_w)*log2e
// Row points P drive the WMMA A fragment; column points Q staged in LDS
// (padded to 4 floats/point for the K=4 B fragment).
// ---------------------------------------------------------------------------
__launch_bounds__(TBLOCK)
__global__ void emd_dual_update(const float* __restrict__ P,
                                const float* __restrict__ Q,
                                const float* __restrict__ normP,
                                const float* __restrict__ normQ,
                                const float* __restrict__ dualIn,
                                float* __restrict__ dualOut,
                                float log_w) {
  __shared__ float q4[NPTS * 4];   // 32 KB: (x,y,z,0) per column point
  __shared__ float t2[NPTS];       //  8 KB: per-column exp2-domain offset

  const int b = blockIdx.y;
  const float* Qb  = Q      + (size_t)b * NPTS * 3;
  const float* nQb = normQ  + (size_t)b * NPTS;
  const float* dIn = dualIn + (size_t)b * NPTS;

  for (int j = threadIdx.x; j < NPTS; j += TBLOCK) {
    float x = Qb[j * 3 + 0], y = Qb[j * 3 + 1], z = Qb[j * 3 + 2];
    q4[j * 4 + 0] = x; q4[j * 4 + 1] = y; q4[j * 4 + 2] = z; q4[j * 4 + 3] = 0.0f;
    t2[j] = ((dIn[j] - nQb[j]) * INV_EPS + log_w) * LOG2E_F;
  }
  __syncthreads();

  const int wave = threadIdx.x >> 5;        // wave32
  const int lane = threadIdx.x & 31;
  const int half = lane >> 4;               // 0: K={0,1} / rows 0-7 ; 1: K={2,3} / rows 8-15
  const int lm   = lane & 15;

  const int rowBase = blockIdx.x * ROWS_PER_BLOCK + wave * ROWS_PER_WAVE;
  const float* Pb = P + ((size_t)b * NPTS + rowBase) * 3;

  // A fragment: 16x4 f32 layout — lanes 0-15 hold K={0,1}, lanes 16-31 K={2,3}
  v2f afrag;
  {
    const float* pr = Pb + lm * 3;
    float ax = (half == 0) ? pr[0] : pr[2];
    float ay = (half == 0) ? pr[1] : 0.0f;
    afrag[0] = ax; afrag[1] = ay;
  }

  // ---------------- Pass A: exact per-row max (no transcendentals) ----------
  float mx[8];
#pragma unroll
  for (int r = 0; r < 8; ++r) mx[r] = -__builtin_inff();

  for (int jc = 0; jc < NPTS; jc += 16) {
    const int j = jc + lm;
    v2f bfrag;
    bfrag[0] = q4[j * 4 + half * 2 + 0];
    bfrag[1] = q4[j * 4 + half * 2 + 1];

    v8f acc = {};
    acc = __builtin_amdgcn_wmma_f32_16x16x4_f32(false, afrag, false, bfrag,
                                                (short)0, acc, false, false);
    const float tj = t2[j];
#pragma unroll
    for (int r = 0; r < 8; ++r)
      mx[r] = fmaxf(mx[r], fmaf(SC2, acc[r], tj));
  }
  // Max across the 16 lanes of each half-wave; all lanes end with the row max.
#pragma unroll
  for (int d = 1; d < 16; d <<= 1) {
#pragma unroll
    for (int r = 0; r < 8; ++r)
      mx[r] = fmaxf(mx[r], __shfl_xor(mx[r], d, 32));
  }

  // ---------------- Pass B: exp2 sum, one transcendental per element --------
  float sm[8];
#pragma unroll
  for (int r = 0; r < 8; ++r) sm[r] = 0.0f;

  for (int jc = 0; jc < NPTS; jc += 16) {
    const int j = jc + lm;
    v2f bfrag;
    bfrag[0] = q4[j * 4 + half * 2 + 0];
    bfrag[1] = q4[j * 4 + half * 2 + 1];

    v8f acc = {};
    acc = __builtin_amdgcn_wmma_f32_16x16x4_f32(false, afrag, false, bfrag,
                                                (short)0, acc, false, false);
    const float tj = t2[j];
#pragma unroll
    for (int r = 0; r < 8; ++r)
      sm[r] += dev_exp2(fmaf(SC2, acc[r], tj) - mx[r]);
  }
  // Sum across the 16 lanes of each half-wave.
#pragma unroll
  for (int d = 1; d < 16; d <<= 1) {
#pragma unroll
    for (int r = 0; r < 8; ++r)
      sm[r] += __shfl_xor(sm[r], d, 32);
  }

  if (lm == 0) {
#pragma unroll
    for (int r = 0; r < 8; ++r) {
      int row = rowBase + half * 8 + r;
      int gi  = b * NPTS + row;
      dualOut[gi] = normP[gi] - (EPS_F * LN2_F) * (mx[r] + __log2f(sm[r]));
    }
  }
}

// ---------------------------------------------------------------------------
// Final cost:  out[b] = N * sum_ij exp((f_i+g_j-C_ij)/eps + log_mu + log_nu) * C_ij
// (evaluated in exp2 domain; C_ij kept in natural units for the multiply)
// ---------------------------------------------------------------------------
__launch_bounds__(TBLOCK)
__global__ void emd_cost(const float* __restrict__ P,
                         const float* __restrict__ Q,
                         const float* __restrict__ normP,
                         const float* __restrict__ normQ,
                         const float* __restrict__ f,
                         const float* __restrict__ g,
                         float* __restrict__ out,
                         float log_muv) {   // log_mu + log_nu (natural log)
  __shared__ float q4[NPTS * 4];   // 32 KB
  __shared__ float h2[NPTS];       // (g_j - b_j)/eps * log2e
  __shared__ float bq[NPTS];       // b_j (natural)

  const int b = blockIdx.y;
  const float* Qb  = Q     + (size_t)b * NPTS * 3;
  const float* nQb = normQ + (size_t)b * NPTS;
  const float* gb  = g     + (size_t)b * NPTS;

  for (int j = threadIdx.x; j < NPTS; j += TBLOCK) {
    float x = Qb[j * 3 + 0], y = Qb[j * 3 + 1], z = Qb[j * 3 + 2];
    q4[j * 4 + 0] = x; q4[j * 4 + 1] = y; q4[j * 4 + 2] = z; q4[j * 4 + 3] = 0.0f;
    float bj = nQb[j];
    bq[j] = bj;
    h2[j] = (gb[j] - bj) * (INV_EPS * LOG2E_F);
  }
  __syncthreads();

  const int wave = threadIdx.x >> 5;
  const int lane = threadIdx.x & 31;
  const int half = lane >> 4;
  const int lm   = lane & 15;

  const int rowBase = blockIdx.x * ROWS_PER_BLOCK + wave * ROWS_PER_WAVE;
  const float* Pb = P + ((size_t)b * NPTS + rowBase) * 3;

  v2f afrag;
  {
    const float* pr = Pb + lm * 3;
    float ax = (half == 0) ? pr[0] : pr[2];
    float ay = (half == 0) ? pr[1] : 0.0f;
    afrag[0] = ax; afrag[1] = ay;
  }

  // Per-row constants for this lane's 8 rows.
  float ai[8], u2[8];
#pragma unroll
  for (int r = 0; r < 8; ++r) {
    int row = rowBase + half * 8 + r;
    int gi  = b * NPTS + row;
    float a = normP[gi];
    ai[r] = a;
    u2[r] = ((f[gi] - a) * INV_EPS + log_muv) * LOG2E_F;
  }

  float accum = 0.0f;
  for (int jc = 0; jc < NPTS; jc += 16) {
    const int j = jc + lm;
    v2f bfrag;
    bfrag[0] = q4[j * 4 + half * 2 + 0];
    bfrag[1] = q4[j * 4 + half * 2 + 1];

    v8f acc = {};
    acc = __builtin_amdgcn_wmma_f32_16x16x4_f32(false, afrag, false, bfrag,
                                                (short)0, acc, false, false);
    const float hj = h2[j];
    const float bj = bq[j];
#pragma unroll
    for (int r = 0; r < 8; ++r) {
      float d    = acc[r];
      float Cij  = ai[r] + bj - 2.0f * d;
      float arg2 = fmaf(SC2, d, u2[r] + hj);
      accum = fmaf(dev_exp2(arg2), Cij, accum);
    }
  }

  // Full-wave reduction, one atomic per wave.
#pragma unroll
  for (int d = 1; d < 32; d <<= 1) accum += __shfl_xor(accum, d, 32);
  if (lane == 0) atomicAdd(&out[b], accum * (float)NPTS);
}

// ---------------------------------------------------------------------------
extern "C" void kernel_launch(void* const* d_in, const int* in_sizes, int n_in,
                              void* d_out, int out_size, void* d_ws, size_t ws_size,
                              hipStream_t stream) {
  (void)in_sizes; (void)n_in; (void)out_size; (void)ws_size;

  const float* p1 = (const float*)d_in[0];
  const float* p2 = (const float*)d_in[1];
  float* out = (float*)d_out;

  float* ws    = (float*)d_ws;
  float* normP = ws;                       // [B*N]
  float* normQ = ws + 1 * BATCH * NPTS;    // [B*N]
  float* f     = ws + 2 * BATCH * NPTS;    // [B*N]
  float* g     = ws + 3 * BATCH * NPTS;    // [B*N]

  const float log_w   = -logf((float)NPTS);       // log_mu == log_nu (N == M)
  const float log_muv = 2.0f * log_w;

  emd_init<<<(BATCH * NPTS + 255) / 256, 256, 0, stream>>>(p1, p2, normP, normQ, f, g, out);

  dim3 grid(NPTS / ROWS_PER_BLOCK, BATCH);  // (32, 8)
  for (int it = 0; it < N_ITERS; ++it) {
    // f <- from g   (rows = p1, cols = p2)
    emd_dual_update<<<grid, TBLOCK, 0, stream>>>(p1, p2, normP, normQ, g, f, log_w);
    // g <- from new f (rows = p2, cols = p1)
    emd_dual_update<<<grid, TBLOCK, 0, stream>>>(p2, p1, normQ, normP, f, g, log_w);
  }

  emd_cost<<<grid, TBLOCK, 0, stream>>>(p1, p2, normP, normQ, f, g, out, log_muv);
}